// radial_conv2d_22316650070871
// MI455X (gfx1250) — compile-verified
//
#include <hip/hip_runtime.h>
#include <math.h>

// CDNA5 / gfx1250, wave32. fp32 WMMA (16x16x4) implicit-GEMM conv with
// double-buffered GLOBAL_LOAD_ASYNC_TO_LDS staging (ASYNCcnt pipelined).

typedef float v2f __attribute__((ext_vector_type(2)));
typedef float v8f __attribute__((ext_vector_type(8)));

#define B_    8
#define S_    8
#define CIN_  64
#define HW_   32
#define COUT_ 128
#define PW_   73728          // COUT*CIN*9
#define P_    73856          // PW + COUT
#define HO_   30
#define WO_   30
#define NTOT_ (B_ * HO_ * WO_)    // 7200
#define KTOT_ (CIN_ * 9)          // 576
#define KC_   16                  // K chunk staged in LDS
#define NCH_  (KTOT_ / KC_)       // 36 chunks
#define NB_   48                  // N columns per block (7200 = 150 * 48)
#define NBLK_ (NTOT_ / NB_)       // 150
#define LDA_  17                  // padded LDS stride for A
#define LDB_  49                  // padded LDS stride for B
#define LDSA_F_ (128 * LDA_)      // floats per A buffer (2176)
#define LDSB_F_ (KC_ * LDB_)      // floats per B buffer (784)
#define OUT_ELEMS_ ((size_t)B_ * S_ * COUT_ * HO_ * WO_)  // 7372800

__device__ __forceinline__ float softplus_f(float x) {
    return log1pf(expf(-fabsf(x))) + fmaxf(x, 0.0f);
}

// Async copy global->LDS (per-lane addresses), tracked by ASYNCcnt.
__device__ __forceinline__ void async_ld_b32(unsigned lds_dst, const float* gsrc) {
    asm volatile("global_load_async_to_lds_b32 %0, %1, off"
                 :: "v"(lds_dst), "v"((unsigned long long)(uintptr_t)gsrc)
                 : "memory");
}
#define WAIT_ASYNCCNT(n) asm volatile("s_wait_asynccnt %0" :: "n"(n) : "memory")

// Low 32 bits of a generic pointer to __shared__ == LDS byte address
// (flat aperture mapping).
__device__ __forceinline__ unsigned lds_off32(const float* p) {
    return (unsigned)(uintptr_t)p;
}

// ---------------------------------------------------------------------------
// Kernel 1: materialize sampled weights w[s,p] and biases b[s,c] into ws.
// ws layout: [0, S*PW) = w ; [S*PW, S*PW + S*COUT) = b
// ---------------------------------------------------------------------------
__global__ void genw_kernel(const float* __restrict__ e,
                            const float* __restrict__ mu_w,
                            const float* __restrict__ rho_w,
                            const float* __restrict__ mu_b,
                            const float* __restrict__ rho_b,
                            float* __restrict__ ws) {
    int i = blockIdx.x * blockDim.x + threadIdx.x;
    const int NW = S_ * PW_;
    if (i < NW) {
        int s = i / PW_;
        int p = i - s * PW_;
        ws[i] = fmaf(softplus_f(rho_w[p]), e[s * P_ + p], mu_w[p]);
    } else if (i < NW + S_ * COUT_) {
        int j = i - NW;
        int s = j / COUT_;
        int c = j - s * COUT_;
        ws[i] = fmaf(softplus_f(rho_b[c]), e[s * P_ + PW_ + c], mu_b[c]);
    }
}

// ---------------------------------------------------------------------------
// Kernel 2: KL scalar. kl = -Σ log σ + log(2π)(PW+COUT)/2 + (Σ w² + Σ b²)/2
// ---------------------------------------------------------------------------
__global__ void kl_kernel(const float* __restrict__ rho_w,
                          const float* __restrict__ rho_b,
                          const float* __restrict__ ws,
                          float* __restrict__ kl_out) {
    __shared__ float s_log[1024];
    __shared__ float s_sq[1024];
    const int tid = threadIdx.x;
    float logsum = 0.0f, sqsum = 0.0f;
    for (int p = tid; p < PW_; p += 1024) logsum += logf(softplus_f(rho_w[p]));
    if (tid < COUT_) logsum += logf(softplus_f(rho_b[tid]));
    const int NALL = S_ * PW_ + S_ * COUT_;
    for (int i = tid; i < NALL; i += 1024) {
        float w = ws[i];
        sqsum = fmaf(w, w, sqsum);
    }
    s_log[tid] = logsum;
    s_sq[tid] = sqsum;
    __syncthreads();
    for (int off = 512; off > 0; off >>= 1) {
        if (tid < off) {
            s_log[tid] += s_log[tid + off];
            s_sq[tid]  += s_sq[tid + off];
        }
        __syncthreads();
    }
    if (tid == 0) {
        const float log2pi = 1.8378770664093453f;
        *kl_out = -s_log[0] + 0.5f * log2pi * (float)(PW_ + COUT_) + 0.5f * s_sq[0];
    }
}

// ---------------------------------------------------------------------------
// Kernel 3: implicit-GEMM conv via V_WMMA_F32_16X16X4_F32 with async-LDS
// double buffering. Grid: (150, 8) blocks of 256 threads (8 wave32).
// Block computes C[128 x 48] for sample blockIdx.y, N-slab blockIdx.x.
// ---------------------------------------------------------------------------
__launch_bounds__(256)
__global__ void conv_wmma_kernel(const float* __restrict__ x,
                                 const float* __restrict__ ws,
                                 float* __restrict__ out) {
    __shared__ float lds_a[2][LDSA_F_];   // 128 x 16 A tiles (w), padded
    __shared__ float lds_b[2][LDSB_F_];   // 16 x 48 B tiles (im2col x), padded

    const int s    = blockIdx.y;
    const int n0   = blockIdx.x * NB_;
    const int tid  = threadIdx.x;
    const int wave = tid >> 5;
    const int lane = tid & 31;
    const int l16  = lane & 15;
    const int khi  = (lane >> 4) << 1;   // 0 for lanes 0-15, 2 for lanes 16-31

    const float* __restrict__ wsmp = ws + (size_t)s * PW_;
    const float* __restrict__ bias = ws + (size_t)S_ * PW_ + s * COUT_;

    // ---- per-thread invariant staging addresses ----
    // A: 2048 elems, 8 per thread. idx = tid + i*256 -> (m, kc)
    const float* ga_base[8];
    unsigned     la_off[8];
    #pragma unroll
    for (int i = 0; i < 8; ++i) {
        int idx = tid + i * 256;
        int m   = idx >> 4;
        int kc  = idx & 15;
        ga_base[i] = wsmp + m * KTOT_ + kc;                 // + k0 per chunk
        la_off[i]  = lds_off32(&lds_a[0][m * LDA_ + kc]);   // + buf*bytes
    }
    // B: 768 elems, 3 per thread. idx = tid + i*256 -> (kc, n)
    const float* gx_base[3];
    unsigned     lb_off[3];
    int          kc_b[3];
    #pragma unroll
    for (int i = 0; i < 3; ++i) {
        int idx = tid + i * 256;
        int kc  = idx / NB_;
        int n   = idx - kc * NB_;
        int gn  = n0 + n;
        int bi  = gn / 900;
        int rem = gn - bi * 900;
        int y   = rem / 30;
        int xw  = rem - y * 30;
        kc_b[i]    = kc;
        gx_base[i] = x + (size_t)((bi * S_ + s) * CIN_) * (HW_ * HW_) + y * HW_ + xw;
        lb_off[i]  = lds_off32(&lds_b[0][kc * LDB_ + n]);
    }

    // Issue the 11 async transfers for chunk with k offset k0v into buffer bufv.
    #define ISSUE_CHUNK(k0v, bufv)                                             \
        do {                                                                   \
            unsigned aoff = (bufv) ? (unsigned)(LDSA_F_ * 4) : 0u;             \
            unsigned boff = (bufv) ? (unsigned)(LDSB_F_ * 4) : 0u;             \
            _Pragma("unroll")                                                  \
            for (int i = 0; i < 8; ++i)                                        \
                async_ld_b32(la_off[i] + aoff, ga_base[i] + (k0v));            \
            _Pragma("unroll")                                                  \
            for (int i = 0; i < 3; ++i) {                                      \
                int kg = (k0v) + kc_b[i];                                      \
                int ci = kg / 9;                                               \
                int r  = kg - ci * 9;                                          \
                int kh = r / 3;                                                \
                int kw = r - kh * 3;                                           \
                async_ld_b32(lb_off[i] + boff,                                 \
                             gx_base[i] + ci * (HW_ * HW_) + kh * HW_ + kw);   \
            }                                                                  \
        } while (0)

    v8f acc0 = {}, acc1 = {}, acc2 = {};

    ISSUE_CHUNK(0, 0);  // prologue: chunk 0 -> buffer 0

    for (int chunk = 0; chunk < NCH_; ++chunk) {
        // Pipeline: issue next chunk, then wait for current chunk's 11
        // transfers (async loads complete in order per wave).
        if (chunk + 1 < NCH_) {
            ISSUE_CHUNK((chunk + 1) * KC_, (chunk + 1) & 1);
            WAIT_ASYNCCNT(11);
        } else {
            WAIT_ASYNCCNT(0);
        }
        __syncthreads();   // all waves' transfers for current buffer visible

        const float* Ab = &lds_a[chunk & 1][0];
        const float* Bb = &lds_b[chunk & 1][0];

        // 4 k-steps of 4, 3 N tiles: 12 fp32 WMMAs per wave per chunk.
        #pragma unroll
        for (int kk = 0; kk < 4; ++kk) {
            v2f a;
            a.x = Ab[(wave * 16 + l16) * LDA_ + 4 * kk + khi];
            a.y = Ab[(wave * 16 + l16) * LDA_ + 4 * kk + khi + 1];
            v2f b0, b1, b2;
            b0.x = Bb[(4 * kk + khi)     * LDB_ +  0 + l16];
            b0.y = Bb[(4 * kk + khi + 1) * LDB_ +  0 + l16];
            b1.x = Bb[(4 * kk + khi)     * LDB_ + 16 + l16];
            b1.y = Bb[(4 * kk + khi + 1) * LDB_ + 16 + l16];
            b2.x = Bb[(4 * kk + khi)     * LDB_ + 32 + l16];
            b2.y = Bb[(4 * kk + khi + 1) * LDB_ + 32 + l16];
            acc0 = __builtin_amdgcn_wmma_f32_16x16x4_f32(
                false, a, false, b0, (short)0, acc0, false, false);
            acc1 = __builtin_amdgcn_wmma_f32_16x16x4_f32(
                false, a, false, b1, (short)0, acc1, false, false);
            acc2 = __builtin_amdgcn_wmma_f32_16x16x4_f32(
                false, a, false, b2, (short)0, acc2, false, false);
        }
        __syncthreads();   // buffer free for re-issue two chunks ahead
    }

    // --- epilogue: bias add + store. C/D layout: VGPR r, lanes 0-15 = row r,
    //     lanes 16-31 = row r+8, N = lane%16.
    #pragma unroll
    for (int r = 0; r < 8; ++r) {
        int co = wave * 16 + r + ((lane >> 4) << 3);
        float bb = bias[co];
        #pragma unroll
        for (int nt = 0; nt < 3; ++nt) {
            float v = (nt == 0 ? acc0[r] : (nt == 1 ? acc1[r] : acc2[r])) + bb;
            int gn  = n0 + nt * 16 + l16;
            int bi  = gn / 900;
            int rem = gn - bi * 900;
            out[(size_t)((bi * S_ + s) * COUT_ + co) * 900 + rem] = v;
        }
    }
    #undef ISSUE_CHUNK
}

// ---------------------------------------------------------------------------
extern "C" void kernel_launch(void* const* d_in, const int* in_sizes, int n_in,
                              void* d_out, int out_size, void* d_ws, size_t ws_size,
                              hipStream_t stream) {
    const float* x     = (const float*)d_in[0];
    const float* e     = (const float*)d_in[1];
    const float* mu_w  = (const float*)d_in[2];
    const float* rho_w = (const float*)d_in[3];
    const float* mu_b  = (const float*)d_in[4];
    const float* rho_b = (const float*)d_in[5];
    float* out = (float*)d_out;
    float* ws  = (float*)d_ws;  // needs (S*PW + S*COUT)*4 = ~2.36 MB

    (void)in_sizes; (void)n_in; (void)out_size; (void)ws_size;

    const int total = S_ * PW_ + S_ * COUT_;
    genw_kernel<<<(total + 255) / 256, 256, 0, stream>>>(e, mu_w, rho_w, mu_b, rho_b, ws);
    kl_kernel<<<1, 1024, 0, stream>>>(rho_w, rho_b, ws, out + OUT_ELEMS_);
    dim3 grid(NBLK_, S_);
    conv_wmma_kernel<<<grid, 256, 0, stream>>>(x, ws, out);
}